// EncoderFeaStConv_80015240725027
// MI455X (gfx1250) — compile-verified
//
#include <hip/hip_runtime.h>

// FeaStConv x2 encoder for MI455X (gfx1250).
// a = h@u and y = h@W precomputed per node -> edge phase is a pure
// gather/blend/scatter that stays in the 192MB L2. GEMMs run on
// v_wmma_f32_16x16x4_f32, one wave = 16x32 output tile (A-frag reused
// across two WMMAs), all strides compile-time so loads use immediate offsets.

typedef float v2f __attribute__((ext_vector_type(2)));
typedef float v8f __attribute__((ext_vector_type(8)));

static constexpr int NNODES = 50000;
static constexpr int NEDGES = 800000;
static constexpr int FIN    = 128;
static constexpr int HID    = 64;
static constexpr int YCH    = 256;   // HEADS * HID

// ---------------------------------------------------------------------------
// GEMM: C[M,N] = A[M,K] @ W[K,N] (+ bias).  K,N compile-time.
// One wave computes a 16(M) x 32(N) tile: two 16x16 WMMA accumulators
// sharing one A fragment per K-step (K stepped by 4).
// A-frag (16x4 f32): lanes 0-15 rows M with K0,K1; lanes 16-31 K2,K3.
// B-frag (4x16 f32): v0 = row {K0|K2}, v1 = row {K1|K3} striped over lanes.
// C/D:  VGPR j -> M = j (lanes 0-15) / 8+j (lanes 16-31), N = lane&15.
// ---------------------------------------------------------------------------
template <int K, int N>
__global__ __launch_bounds__(256)
void gemm_wmma_kernel(const float* __restrict__ A,
                      const float* __restrict__ W,
                      const float* __restrict__ bias,
                      float* __restrict__ C,
                      int M, int totalWork) {
  const int wave = threadIdx.x >> 5;
  const int lane = threadIdx.x & 31;
  const int work = blockIdx.x * 8 + wave;
  if (work >= totalWork) return;             // wave-uniform -> EXEC all 1s
  const int mtiles = M >> 4;
  const int npair  = work / mtiles;          // which 32-wide N panel
  const int mtile  = work - npair * mtiles;

  const int half  = lane >> 4;               // 0 | 1
  const int l15   = lane & 15;
  const int khalf = half << 1;               // 0 | 2

  const unsigned m  = ((unsigned)mtile << 4) + (unsigned)l15;
  const unsigned n0 = ((unsigned)npair << 5) + (unsigned)l15;

  const float* ap  = A + m * (unsigned)K + (unsigned)khalf;  // A row base
  const float* wb0 = W + (unsigned)khalf * N + n0;           // B col base
  const float* wb1 = wb0 + 16;

  v8f acc0 = {0.f, 0.f, 0.f, 0.f, 0.f, 0.f, 0.f, 0.f};
  v8f acc1 = {0.f, 0.f, 0.f, 0.f, 0.f, 0.f, 0.f, 0.f};

#pragma unroll
  for (int k = 0; k < K; k += 4) {
    const float2 a2 = *(const float2*)(ap + k);   // b64, immediate offset
    v2f af; af.x = a2.x; af.y = a2.y;
    v2f b0; b0.x = wb0[k * N]; b0.y = wb0[(k + 1) * N];   // imm offsets
    v2f b1; b1.x = wb1[k * N]; b1.y = wb1[(k + 1) * N];
    acc0 = __builtin_amdgcn_wmma_f32_16x16x4_f32(false, af, false, b0,
                                                 (short)0, acc0, false, false);
    acc1 = __builtin_amdgcn_wmma_f32_16x16x4_f32(false, af, false, b1,
                                                 (short)0, acc1, false, false);
  }

  const float bv0 = bias ? bias[n0] : 0.0f;
  const float bv1 = bias ? bias[n0 + 16] : 0.0f;
  const unsigned mrow = ((unsigned)mtile << 4) + ((unsigned)half << 3);
  float* c0 = C + mrow * (unsigned)N + n0;
#pragma unroll
  for (int j = 0; j < 8; ++j) {
    c0[(unsigned)(j * N)]      = acc0[j] + bv0;
    c0[(unsigned)(j * N) + 16] = acc1[j] + bv1;
  }
}

// a[n,h] = dot(H[n,:], U[:,h])  (H:[M,64], U:[64,4]) — one thread per node.
__global__ __launch_bounds__(256)
void attn_proj_kernel(const float* __restrict__ H,
                      const float* __restrict__ U,
                      float* __restrict__ Aout, int M) {
  int n = blockIdx.x * blockDim.x + threadIdx.x;
  if (n >= M) return;
  float4 acc = {0.f, 0.f, 0.f, 0.f};
  const float* hrow = H + (unsigned)n * HID;
#pragma unroll 8
  for (int k = 0; k < HID; ++k) {
    const float  hv = hrow[k];
    const float4 uv = *(const float4*)(U + k * 4);
    acc.x += hv * uv.x; acc.y += hv * uv.y;
    acc.z += hv * uv.z; acc.w += hv * uv.w;
  }
  *(float4*)(Aout + (unsigned)n * 4) = acc;
}

// In-degree (same for both layers): cnt[dst] += 1.
__global__ __launch_bounds__(256)
void degree_kernel(const long long* __restrict__ ei, float* __restrict__ cnt,
                   int E) {
  int e = blockIdx.x * blockDim.x + threadIdx.x;
  if (e >= E) return;
  int dst = (int)ei[(unsigned)E + (unsigned)e];
  unsafeAtomicAdd(&cnt[dst], 1.0f);
}

// Edge phase: 16 lanes per edge, lane handles one channel quad.
// q = softmax(a[src]-a[dst]+c); msg = sum_h q_h * y[src, h*64+d];
// agg[dst] += msg via global_atomic_add_f32 (L2-resident).
__global__ __launch_bounds__(256)
void edge_kernel(const float* __restrict__ Y,    // [M,256]
                 const float* __restrict__ Aq,   // [M,4]
                 const float* __restrict__ Cv,   // [4]
                 const long long* __restrict__ ei,
                 float* __restrict__ Agg,        // [M,64]
                 int E) {
  int t = blockIdx.x * blockDim.x + threadIdx.x;
  int e = t >> 4;
  if (e >= E) return;
  const unsigned sub = ((unsigned)t & 15u) << 2;   // channel quad base
  const unsigned j = (unsigned)(int)ei[e];                        // src
  const unsigned i = (unsigned)(int)ei[(unsigned)E + (unsigned)e]; // dst

  const float4 aj = *(const float4*)(Aq + j * 4u);
  const float4 ai = *(const float4*)(Aq + i * 4u);
  const float4 cv = *(const float4*)(Cv);
  float t0 = aj.x - ai.x + cv.x, t1 = aj.y - ai.y + cv.y;
  float t2 = aj.z - ai.z + cv.z, t3 = aj.w - ai.w + cv.w;
  const float mx = fmaxf(fmaxf(t0, t1), fmaxf(t2, t3));
  const float e0 = __expf(t0 - mx), e1 = __expf(t1 - mx);
  const float e2 = __expf(t2 - mx), e3 = __expf(t3 - mx);
  const float inv = 1.0f / (e0 + e1 + e2 + e3);
  const float q0 = e0 * inv, q1 = e1 * inv, q2 = e2 * inv, q3 = e3 * inv;

  const float* yb = Y + j * (unsigned)YCH + sub;
  const float4 y0 = *(const float4*)(yb);
  const float4 y1 = *(const float4*)(yb + 64);
  const float4 y2 = *(const float4*)(yb + 128);
  const float4 y3 = *(const float4*)(yb + 192);

  float4 msg;
  msg.x = q0 * y0.x + q1 * y1.x + q2 * y2.x + q3 * y3.x;
  msg.y = q0 * y0.y + q1 * y1.y + q2 * y2.y + q3 * y3.y;
  msg.z = q0 * y0.z + q1 * y1.z + q2 * y2.z + q3 * y3.z;
  msg.w = q0 * y0.w + q1 * y1.w + q2 * y2.w + q3 * y3.w;

  float* d = Agg + i * (unsigned)HID + sub;
  unsafeAtomicAdd(d + 0, msg.x);
  unsafeAtomicAdd(d + 1, msg.y);
  unsafeAtomicAdd(d + 2, msg.z);
  unsafeAtomicAdd(d + 3, msg.w);
}

// out = relu(agg/max(cnt,1) + bias + res) — 16 threads per node (float4 each).
__global__ __launch_bounds__(256)
void finalize_kernel(const float* __restrict__ Agg,
                     const float* __restrict__ Cnt,
                     const float* __restrict__ bias,
                     const float* __restrict__ Res,
                     float* __restrict__ Out, int M) {
  int t = blockIdx.x * blockDim.x + threadIdx.x;
  int n = t >> 4;
  if (n >= M) return;
  const unsigned sub = ((unsigned)t & 15u) << 2;
  const float invc = 1.0f / fmaxf(Cnt[n], 1.0f);
  const unsigned off = (unsigned)n * HID + sub;
  const float4 a = *(const float4*)(Agg + off);
  const float4 r = *(const float4*)(Res + off);
  const float4 b = *(const float4*)(bias + sub);
  float4 o;
  o.x = fmaxf(a.x * invc + b.x + r.x, 0.0f);
  o.y = fmaxf(a.y * invc + b.y + r.y, 0.0f);
  o.z = fmaxf(a.z * invc + b.z + r.z, 0.0f);
  o.w = fmaxf(a.w * invc + b.w + r.w, 0.0f);
  *(float4*)(Out + off) = o;
}

extern "C" void kernel_launch(void* const* d_in, const int* in_sizes, int n_in,
                              void* d_out, int out_size, void* d_ws, size_t ws_size,
                              hipStream_t stream) {
  (void)in_sizes; (void)n_in; (void)out_size; (void)ws_size;

  const float*     x       = (const float*)d_in[0];
  const long long* ei      = (const long long*)d_in[1];   // jnp.int64
  const float*     lin1_w  = (const float*)d_in[2];
  const float*     lin1_b  = (const float*)d_in[3];
  const float*     lin2_w  = (const float*)d_in[4];
  const float*     lin2_b  = (const float*)d_in[5];
  const float*     lin3_w  = (const float*)d_in[6];
  const float*     lin3_b  = (const float*)d_in[7];
  const float*     c1_w    = (const float*)d_in[8];
  const float*     c1_u    = (const float*)d_in[9];
  const float*     c1_c    = (const float*)d_in[10];
  const float*     c1_bias = (const float*)d_in[11];
  const float*     c2_w    = (const float*)d_in[12];
  const float*     c2_u    = (const float*)d_in[13];
  const float*     c2_c    = (const float*)d_in[14];
  const float*     c2_bias = (const float*)d_in[15];

  const int M = NNODES, E = NEDGES;

  // Workspace layout (floats). h1 buffer is reused as agg once y/a are built.
  float* ws    = (float*)d_ws;
  float* h1agg = ws;                        // [M,64]  h1, then agg
  float* res   = ws + (size_t)M * HID;      // [M,64]  x@lin2 then h@lin3
  float* Y     = res + (size_t)M * HID;     // [M,256]
  float* Aq    = Y + (size_t)M * YCH;       // [M,4]
  float* cnt   = Aq + (size_t)M * 4;        // [M]
  float* H     = cnt + M;                   // [M,64]  layer-1 output

  const dim3 blk(256);
  // one wave = 16x32 tile; work items = (M/16) * (N/32); 8 waves per block
  const int w64  = (M >> 4) * (HID >> 5);   // 6250
  const int w256 = (M >> 4) * (YCH >> 5);   // 25000
  const dim3 g64((unsigned)((w64 + 7) / 8));
  const dim3 g256((unsigned)((w256 + 7) / 8));

  const dim3 nodeGrid((M + 255) / 256);                 // 1 thread / node
  const dim3 edgeGrid((E + 255) / 256);                 // 1 thread / edge
  const dim3 edge16Grid((unsigned)(((size_t)E * 16 + 255) / 256));
  const dim3 node16Grid((unsigned)(((size_t)M * 16 + 255) / 256));

  // ---- layer 1 ----
  gemm_wmma_kernel<FIN, HID><<<g64, blk, 0, stream>>>(x, lin1_w, lin1_b,
                                                      h1agg, M, w64);
  gemm_wmma_kernel<FIN, HID><<<g64, blk, 0, stream>>>(x, lin2_w, lin2_b,
                                                      res, M, w64);
  gemm_wmma_kernel<HID, YCH><<<g256, blk, 0, stream>>>(h1agg, c1_w, nullptr,
                                                       Y, M, w256);
  attn_proj_kernel<<<nodeGrid, blk, 0, stream>>>(h1agg, c1_u, Aq, M);
  hipMemsetAsync(cnt, 0, (size_t)M * sizeof(float), stream);
  degree_kernel<<<edgeGrid, blk, 0, stream>>>(ei, cnt, E);
  hipMemsetAsync(h1agg, 0, (size_t)M * HID * sizeof(float), stream);
  edge_kernel<<<edge16Grid, blk, 0, stream>>>(Y, Aq, c1_c, ei, h1agg, E);
  finalize_kernel<<<node16Grid, blk, 0, stream>>>(h1agg, cnt, c1_bias, res, H, M);

  // ---- layer 2 ----
  gemm_wmma_kernel<HID, YCH><<<g256, blk, 0, stream>>>(H, c2_w, nullptr,
                                                       Y, M, w256);
  attn_proj_kernel<<<nodeGrid, blk, 0, stream>>>(H, c2_u, Aq, M);
  gemm_wmma_kernel<HID, HID><<<g64, blk, 0, stream>>>(H, lin3_w, lin3_b,
                                                      res, M, w64);
  hipMemsetAsync(h1agg, 0, (size_t)M * HID * sizeof(float), stream);
  edge_kernel<<<edge16Grid, blk, 0, stream>>>(Y, Aq, c2_c, ei, h1agg, E);
  finalize_kernel<<<node16Grid, blk, 0, stream>>>(h1agg, cnt, c2_bias, res,
                                                  (float*)d_out, M);
}